// HyenaFilter_58420145160627
// MI455X (gfx1250) — compile-verified
//
#include <hip/hip_runtime.h>

typedef __attribute__((ext_vector_type(2))) float v2f;
typedef __attribute__((ext_vector_type(4))) float v4f;
typedef __attribute__((ext_vector_type(8))) float v8f;
typedef __attribute__((ext_vector_type(4))) int   v4i;

#define BATCH   4
#define SEQ     8192
#define DIM     1024
#define LFILT   2048

#define TM      64          // time rows per workgroup
#define TN      128         // dim cols per workgroup (8 waves x 16)
#define KC      32          // K (s) rows staged per chunk
#define TNP     136         // padded LDS row stride (bank-conflict avoidance)
#define PAD_LO  64
#define FILT_LDS (PAD_LO + LFILT + 64)           // 2176 floats
#define NCHUNK  ((LFILT + TM + KC - 1) / KC)     // 66 chunks -> g in [0,2112)

#define AS_GLOBAL __attribute__((address_space(1)))
#define AS_LDS    __attribute__((address_space(3)))

__global__ __launch_bounds__(256)
void hyena_conv_wmma(const float* __restrict__ x,
                     const float* __restrict__ filt,
                     float* __restrict__ out)
{
    __shared__ float sfilt[FILT_LDS];
    __shared__ float sx[2][KC][TNP];

    const int tid  = threadIdx.x;
    const int lane = tid & 31;
    const int wave = tid >> 5;
    const int ml   = lane & 15;   // M (A/C row sub-index) / N column
    const int kh   = lane >> 4;   // half-wave: K pair selector / M high bit

    // --- block decomposition: 128 time-tiles x 8 dim-tiles x 4 batches ---
    const int nTM = SEQ / TM;     // 128
    const int nTN = DIM / TN;     // 8
    int bid  = blockIdx.x;
    const int tIdx = bid % nTM;
    int rest = bid / nTM;
    const int dIdx = rest % nTN;
    const int b    = rest / nTN;

    const int t0 = tIdx * TM;
    const int d0 = dIdx * TN;

    // --- zero-padded filter in LDS: sfilt[PAD_LO + j] = filt[j], zeros outside ---
    for (int i = tid; i < FILT_LDS; i += 256) {
        int fi = i - PAD_LO;
        sfilt[i] = (fi >= 0 && fi < LFILT) ? filt[fi] : 0.0f;
    }

    const float* xb = x + (size_t)b * SEQ * DIM;

    // this thread's staging slots: 4 x float4 per chunk (32 rows x 128 floats)
    // f = tid + k*256; row = f>>5; c4 = (f&31)*4  -> lanes 0..31 cover a row contiguously

    // --- stage chunk 0 via async DMA to LDS ---
    {
        const int sbase = t0;
        #pragma unroll
        for (int k = 0; k < 4; ++k) {
            int f   = tid + k * 256;
            int row = f >> 5;
            int c4  = (f & 31) << 2;
            int s   = sbase + row;
            float* ldst = &sx[0][row][c4];
            if (s < SEQ) {
                __builtin_amdgcn_global_load_async_to_lds_b128(
                    (AS_GLOBAL v4i*)(xb + (size_t)s * DIM + d0 + c4),
                    (AS_LDS v4i*)ldst, 0, 0);
            } else {
                *(v4f*)ldst = (v4f){0.f, 0.f, 0.f, 0.f};
            }
        }
        asm volatile("s_wait_asynccnt 0" ::: "memory");
    }
    __syncthreads();

    v8f acc[4];
    #pragma unroll
    for (int i = 0; i < 4; ++i) acc[i] = (v8f){0.f,0.f,0.f,0.f,0.f,0.f,0.f,0.f};

    const int dcolL = wave * 16 + ml;   // this lane's column within the LDS slab

    for (int ch = 0; ch < NCHUNK; ++ch) {
        const int buf = ch & 1;

        // kick off async DMA for the next chunk; WMMAs below hide the latency
        if (ch + 1 < NCHUNK) {
            const int sbase = t0 + (ch + 1) * KC;
            #pragma unroll
            for (int k = 0; k < 4; ++k) {
                int f   = tid + k * 256;
                int row = f >> 5;
                int c4  = (f & 31) << 2;
                int s   = sbase + row;
                float* ldst = &sx[buf ^ 1][row][c4];
                if (s < SEQ) {
                    __builtin_amdgcn_global_load_async_to_lds_b128(
                        (AS_GLOBAL v4i*)(xb + (size_t)s * DIM + d0 + c4),
                        (AS_LDS v4i*)ldst, 0, 0);
                } else {
                    *(v4f*)ldst = (v4f){0.f, 0.f, 0.f, 0.f};  // tail padding
                }
            }
        }

        // --- 8 K-steps of 4 over this chunk: 32 WMMAs per wave ---
        #pragma unroll
        for (int ks = 0; ks < 8; ++ks) {
            const int g  = ch * KC + 4 * ks;     // absolute K offset (s - t0)
            const int r0 = 4 * ks + 2 * kh;      // B rows for this half-wave

            v2f bv;
            bv.x = sx[buf][r0    ][dcolL];       // B: K = 2*kh
            bv.y = sx[buf][r0 + 1][dcolL];       // B: K = 2*kh + 1

            const int abase = PAD_LO + g + 2 * kh - ml;
            #pragma unroll
            for (int i = 0; i < 4; ++i) {
                v2f av;                           // A: Toeplitz filt[g + k - 16i - m]
                av.x = sfilt[abase - 16 * i];
                av.y = sfilt[abase - 16 * i + 1];
                acc[i] = __builtin_amdgcn_wmma_f32_16x16x4_f32(
                    false, av, false, bv, (short)0, acc[i], false, false);
            }
        }

        if (ch + 1 < NCHUNK) {
            asm volatile("s_wait_asynccnt 0" ::: "memory");
        }
        __syncthreads();
    }

    // --- write out: C/D layout: VGPR r -> M = r + 8*kh, N = ml ---
    const int dcol = d0 + dcolL;
    float* ob = out + (size_t)b * SEQ * DIM + dcol;
    #pragma unroll
    for (int i = 0; i < 4; ++i) {
        #pragma unroll
        for (int r = 0; r < 8; ++r) {
            int t = t0 + 16 * i + r + 8 * kh;
            ob[(size_t)t * DIM] = acc[i][r];
        }
    }
}

extern "C" void kernel_launch(void* const* d_in, const int* in_sizes, int n_in,
                              void* d_out, int out_size, void* d_ws, size_t ws_size,
                              hipStream_t stream) {
    const float* x    = (const float*)d_in[0];
    const float* filt = (const float*)d_in[1];
    float* out        = (float*)d_out;

    dim3 grid((SEQ / TM) * (DIM / TN) * BATCH);  // 128 * 8 * 4 = 4096
    dim3 block(256);
    hyena_conv_wmma<<<grid, block, 0, stream>>>(x, filt, out);
}